// eKDS_45741401702955
// MI455X (gfx1250) — compile-verified
//
#include <hip/hip_runtime.h>
#include <hip/hip_bf16.h>

// ---------------- Problem constants (from reference) ----------------
#define NI   6144
#define NJ   6144
#define MM   1024      // rows of f
#define KK   256       // cols of f
#define COLS 1024      // columns per block chunk (kernel 1)
#define ROWS 16        // rows per block (kernel 1)
#define NCHUNK (NJ / COLS)      // 6
#define NRGRP  (NI / ROWS)      // 384
#define NPART  (NCHUNK * NRGRP) // 2304 partial sums
#define INV_R  1.25f            // 1/0.8

typedef __attribute__((ext_vector_type(2))) float v2f;
typedef __attribute__((ext_vector_type(8))) float v8f;

// =====================================================================
// Kernel 1: streaming EMD partial sums.  acc += (|dy0|+|dy1|) * x_ij
// 256 threads/block.  Each block: 16 rows x 1024 cols of x_ij.
// y_j chunk staged through LDS once, then held in registers.
// =====================================================================
__global__ void __launch_bounds__(256)
emd_partial_kernel(const float* __restrict__ yi_g,
                   const float* __restrict__ yj_g,
                   const float* __restrict__ x,
                   float* __restrict__ partials) {
    __shared__ float2 sh_yj[COLS];
    __shared__ float  red[256];

    const int tid     = threadIdx.x;
    const int colBase = blockIdx.x * COLS;
    const int rowBase = blockIdx.y * ROWS;

    const float2* yj2 = (const float2*)yj_g;  // y_j is [NJ][2] f32
    for (int t = tid; t < COLS; t += 256) sh_yj[t] = yj2[colBase + t];
    __syncthreads();

    // This thread's 4 fixed column positions (reused across all 16 rows)
    const float2 ya0 = sh_yj[tid * 4 + 0];
    const float2 ya1 = sh_yj[tid * 4 + 1];
    const float2 ya2 = sh_yj[tid * 4 + 2];
    const float2 ya3 = sh_yj[tid * 4 + 3];

    const float2* yi2 = (const float2*)yi_g;
    float acc = 0.0f;

    for (int r = 0; r < ROWS; ++r) {
        const int i = rowBase + r;
        // prefetch next row's cacheline for this lane (global_prefetch_b8)
        if (r + 1 < ROWS)
            __builtin_prefetch(x + (size_t)(i + 1) * NJ + colBase + tid * 4, 0, 0);

        const float2 yi = yi2[i];  // uniform per block-row -> scalar load
        const float4* xr = (const float4*)(x + (size_t)i * NJ + colBase);
        const float4 xv = xr[tid];  // coalesced global_load_b128

        // theta*x with abs folded into the add (abs src modifiers), then fma
        acc = fmaf(fabsf(yi.x - ya0.x) + fabsf(yi.y - ya0.y), xv.x, acc);
        acc = fmaf(fabsf(yi.x - ya1.x) + fabsf(yi.y - ya1.y), xv.y, acc);
        acc = fmaf(fabsf(yi.x - ya2.x) + fabsf(yi.y - ya2.y), xv.z, acc);
        acc = fmaf(fabsf(yi.x - ya3.x) + fabsf(yi.y - ya3.y), xv.w, acc);
    }

    // deterministic block tree-reduction
    red[tid] = acc;
    __syncthreads();
    for (int s = 128; s > 0; s >>= 1) {
        if (tid < s) red[tid] += red[tid + s];
        __syncthreads();
    }
    if (tid == 0) partials[blockIdx.y * NCHUNK + blockIdx.x] = red[0];
}

// =====================================================================
// Kernel 2: deterministic reduction of the 2304 partials -> d_out[0]
// =====================================================================
__global__ void __launch_bounds__(256)
emd_reduce_kernel(const float* __restrict__ partials, float* __restrict__ out) {
    __shared__ float red[256];
    float s = 0.0f;
    for (int i = threadIdx.x; i < NPART; i += 256) s += partials[i];
    red[threadIdx.x] = s;
    __syncthreads();
    for (int k = 128; k > 0; k >>= 1) {
        if (threadIdx.x < k) red[threadIdx.x] += red[threadIdx.x + k];
        __syncthreads();
    }
    if (threadIdx.x == 0) out[0] = red[0] * INV_R;  // fold the /R once
}

// =====================================================================
// Kernel 3: per-row simplex projection.  One wave (32 lanes) per row.
//   1) bitonic sort (descending) of 256 elems in LDS
//   2) cumsum via triangular matmul on V_WMMA_F32_16X16X4_F32:
//        row as 16x16 tile U;  P = U x T,  T[k][n] = (k <= n)
//      (4 chained K=4 WMMA accumulations), plus 16 segment offsets
//   3) rho = #{u > v} - 1 (wave reduction), w = v[rho], out = relu(f - w)
// =====================================================================
__global__ void __launch_bounds__(32)
simplex_project_kernel(const float* __restrict__ f, float* __restrict__ z) {
    __shared__ float sh_u[KK];
    __shared__ float sh_v[KK];
    __shared__ float sh_tot[16];

    const int lane = threadIdx.x;        // 0..31
    const int row  = blockIdx.x;         // 0..1023
    const int h    = lane >> 4;          // lane half
    const int n    = lane & 15;          // column / M-row index

    const float* frow = f + (size_t)row * KK;

    for (int t = lane; t < KK; t += 32) sh_u[t] = frow[t];
    __syncthreads();

    // ---- bitonic sort, descending ----
    for (int k = 2; k <= KK; k <<= 1) {
        for (int j = k >> 1; j > 0; j >>= 1) {
            for (int t = lane; t < KK; t += 32) {
                const int ixj = t ^ j;
                if (ixj > t) {
                    const float a = sh_u[t];
                    const float b = sh_u[ixj];
                    const bool desc = ((t & k) == 0);
                    if (desc ? (a < b) : (a > b)) {
                        sh_u[t]   = b;
                        sh_u[ixj] = a;
                    }
                }
            }
            __syncthreads();
        }
    }

    // ---- cumsum via WMMA:  D[m][n] = sum_k U[m][k] * (k <= n) ----
    // A layout (16x4 f32): lane = M (mod 16), K = 2*(lane/16) + vgpr
    // B layout (4x16 f32): K = 2*(lane/16) + vgpr, N = lane%16
    v8f acc = {};
    #pragma unroll
    for (int kk = 0; kk < 4; ++kk) {
        v2f a, b;
        const int kbase = kk * 4 + 2 * h;
        a.x = sh_u[n * 16 + kbase + 0];     // U[m=n][kbase+0]
        a.y = sh_u[n * 16 + kbase + 1];
        b.x = ((kbase + 0) <= n) ? 1.0f : 0.0f;   // upper-triangular ones
        b.y = ((kbase + 1) <= n) ? 1.0f : 0.0f;
        acc = __builtin_amdgcn_wmma_f32_16x16x4_f32(
            /*neg_a=*/false, a, /*neg_b=*/false, b,
            /*c_mod=*/(short)0, acc, /*reuse_a=*/false, /*reuse_b=*/false);
    }

    // segment totals: D[m][15] held by lanes with n == 15 (vgpr v -> m = 8h+v)
    if (n == 15) {
        #pragma unroll
        for (int v = 0; v < 8; ++v) sh_tot[8 * h + v] = acc[v];
    }
    __syncthreads();

    // exclusive prefix of the 16 segment totals (each lane keeps its 8)
    float off[8];
    float run = 0.0f;
    #pragma unroll
    for (int m = 0; m < 16; ++m) {
        if ((m >> 3) == h) off[m & 7] = run;
        run += sh_tot[m];
    }

    // v_t = (cumsum_t - 1)/(t+1);  count indicator u_t > v_t
    int cnt = 0;
    #pragma unroll
    for (int v = 0; v < 8; ++v) {
        const int m    = 8 * h + v;
        const int flat = m * 16 + n;
        const float cs   = acc[v] + off[v];
        const float vval = (cs - 1.0f) / (float)(flat + 1);
        sh_v[flat] = vval;
        if (sh_u[flat] > vval) cnt++;
    }
    __syncthreads();

    // wave32 butterfly sum -> rho
    #pragma unroll
    for (int s = 16; s > 0; s >>= 1) cnt += __shfl_xor(cnt, s, 32);
    const int rho = cnt - 1;
    const float w = sh_v[rho];

    // z = relu(f - w) against the ORIGINAL (unsorted) row
    float* zrow = z + (size_t)row * KK;
    for (int t = lane; t < KK; t += 32) {
        const float d = frow[t] - w;
        zrow[t] = d > 0.0f ? d : 0.0f;
    }
}

// =====================================================================
extern "C" void kernel_launch(void* const* d_in, const int* in_sizes, int n_in,
                              void* d_out, int out_size, void* d_ws, size_t ws_size,
                              hipStream_t stream) {
    const float* y_i  = (const float*)d_in[0];   // [6144, 2]
    const float* y_j  = (const float*)d_in[1];   // [6144, 2]
    const float* x_ij = (const float*)d_in[2];   // [6144, 6144]
    const float* f    = (const float*)d_in[3];   // [1024, 256]
    float* out      = (float*)d_out;             // [1 + 1024*256]
    float* partials = (float*)d_ws;              // NPART floats of scratch

    emd_partial_kernel<<<dim3(NCHUNK, NRGRP), 256, 0, stream>>>(y_i, y_j, x_ij, partials);
    emd_reduce_kernel<<<1, 256, 0, stream>>>(partials, out);
    simplex_project_kernel<<<MM, 32, 0, stream>>>(f, out + 1);
}